// GNN_50002009260728
// MI455X (gfx1250) — compile-verified
//
#include <hip/hip_runtime.h>
#include <hip/hip_bf16.h>
#include <math.h>

// Problem constants (match reference)
constexpr int N_NODES = 100000;
constexpr int N_EDGES = 1600000;
constexpr int EP      = N_EDGES + N_NODES;   // edges + self loops
constexpr int NGRAPH  = 512;
constexpr int DH      = 128;
constexpr float NEG_SLOPE = 0.2f;

typedef __attribute__((ext_vector_type(2))) float v2f;
typedef __attribute__((ext_vector_type(8))) float v8f;

// ---------- float <-> order-preserving uint (for atomicMax on floats) ----------
__device__ __forceinline__ unsigned f2ord(float f) {
  unsigned u = __float_as_uint(f);
  return (u & 0x80000000u) ? ~u : (u | 0x80000000u);
}
__device__ __forceinline__ float ord2f(unsigned u) {
  u = (u & 0x80000000u) ? (u & 0x7FFFFFFFu) : ~u;
  return __uint_as_float(u);
}

// ---------- generic fill ----------
__global__ void fill_u32_kernel(unsigned* __restrict__ p, unsigned val, int count) {
  int i = blockIdx.x * blockDim.x + threadIdx.x;
  if (i < count) p[i] = val;
}

// ---------- GEMM: H[n,128] = X[n,128] @ W[128,128] via V_WMMA_F32_16X16X4_F32 ----------
// Block: 256 threads = 8 waves. Each wave -> 16x16 output tile.
// Block covers 128 rows (wave w -> rows blockIdx.x*128 + w*16) x 16 cols (blockIdx.y*16).
// LDS caches the 16 needed W columns: sW[k][c] , k=0..127, c=0..15  (8KB).
__global__ __launch_bounds__(256)
void gemm_wmma_kernel(const float* __restrict__ X, const float* __restrict__ W,
                      float* __restrict__ H, int n) {
  __shared__ float sW[DH * 16];

  const int tid  = threadIdx.x;
  const int col0 = blockIdx.y * 16;

  // cooperative load of W[:, col0:col0+16] into LDS
  for (int idx = tid; idx < DH * 16; idx += 256) {
    int k = idx >> 4, c = idx & 15;
    sW[idx] = W[k * DH + col0 + c];
  }
  __syncthreads();

  const int wave  = tid >> 5;
  const int lane  = tid & 31;
  const int lhalf = lane >> 4;   // 0: lanes 0-15, 1: lanes 16-31
  const int l16   = lane & 15;

  const int row0 = blockIdx.x * 128 + wave * 16;
  if (row0 >= n) return;  // whole wave exits together (EXEC stays all-ones otherwise)

  int arow = row0 + l16;
  if (arow >= n) arow = n - 1;               // clamp reads for partial tile
  const float* aptr = X + (size_t)arow * DH;

  v8f acc = {};
  #pragma unroll 8
  for (int k = 0; k < DH; k += 4) {
    // A fragment (16x4 f32): vgpr0 = K = k + 2*lhalf, vgpr1 = K+1
    v2f a;
    a.x = aptr[k + lhalf * 2 + 0];
    a.y = aptr[k + lhalf * 2 + 1];
    // B fragment (4x16 f32): vgpr0 = B[k + 2*lhalf][col0 + l16], vgpr1 = next K
    v2f b;
    b.x = sW[(k + lhalf * 2 + 0) * 16 + l16];
    b.y = sW[(k + lhalf * 2 + 1) * 16 + l16];
    acc = __builtin_amdgcn_wmma_f32_16x16x4_f32(false, a, false, b,
                                                (short)0, acc, false, false);
  }

  // C/D layout: vgpr r -> M = r + 8*lhalf, N = l16
  const int ocol = col0 + l16;
  #pragma unroll
  for (int r = 0; r < 8; ++r) {
    int orow = row0 + lhalf * 8 + r;
    if (orow < n) H[(size_t)orow * DH + ocol] = acc[r];
  }
}

// ---------- per-node attention scores: s = <h, a> (wave per node) ----------
__global__ __launch_bounds__(256)
void score_kernel(const float* __restrict__ H, const float* __restrict__ a_src,
                  const float* __restrict__ a_dst, float* __restrict__ s_src,
                  float* __restrict__ s_dst, int n) {
  int node = blockIdx.x * 8 + (threadIdx.x >> 5);
  if (node >= n) return;
  int lane = threadIdx.x & 31;
  float4 h  = ((const float4*)(H + (size_t)node * DH))[lane];
  float4 as = ((const float4*)a_src)[lane];
  float4 ad = ((const float4*)a_dst)[lane];
  float ss = h.x*as.x + h.y*as.y + h.z*as.z + h.w*as.w;
  float sd = h.x*ad.x + h.y*ad.y + h.z*ad.z + h.w*ad.w;
  #pragma unroll
  for (int off = 16; off > 0; off >>= 1) {
    ss += __shfl_xor(ss, off, 32);
    sd += __shfl_xor(sd, off, 32);
  }
  if (lane == 0) { s_src[node] = ss; s_dst[node] = sd; }
}

__device__ __forceinline__ void edge_sd(const int* __restrict__ ei, int i, int& s, int& d) {
  if (i < N_EDGES) { s = ei[i]; d = ei[N_EDGES + i]; }
  else             { s = d = i - N_EDGES; }   // self-loop
}

// ---------- edge score + segment max over dst ----------
__global__ void edge_max_kernel(const int* __restrict__ ei, const float* __restrict__ s_src,
                                const float* __restrict__ s_dst, float* __restrict__ e,
                                unsigned* __restrict__ mEnc) {
  int i = blockIdx.x * blockDim.x + threadIdx.x;
  if (i >= EP) return;
  int s, d; edge_sd(ei, i, s, d);
  float v = s_src[s] + s_dst[d];
  v = v > 0.f ? v : NEG_SLOPE * v;             // leaky_relu
  e[i] = v;
  atomicMax(&mEnc[d], f2ord(v));
}

// ---------- exp(e - m[dst]) + segment sum denom ----------
__global__ void edge_exp_kernel(const int* __restrict__ ei, float* __restrict__ e,
                                const unsigned* __restrict__ mEnc, float* __restrict__ denom) {
  int i = blockIdx.x * blockDim.x + threadIdx.x;
  if (i >= EP) return;
  int s, d; edge_sd(ei, i, s, d);
  float ex = expf(e[i] - ord2f(mEnc[d]));
  e[i] = ex;
  atomicAdd(&denom[d], ex);
}

// ---------- weighted scatter-add: agg[dst] += alpha * h[src] (wave per edge) ----------
__global__ __launch_bounds__(256)
void edge_aggr_kernel(const int* __restrict__ ei, const float* __restrict__ e,
                      const float* __restrict__ denom, const float* __restrict__ H,
                      float* __restrict__ agg) {
  int edge = blockIdx.x * 8 + (threadIdx.x >> 5);
  if (edge >= EP) return;
  int lane = threadIdx.x & 31;
  int s, d; edge_sd(ei, edge, s, d);
  float alpha = e[edge] / denom[d];
  float4 hv = ((const float4*)(H + (size_t)s * DH))[lane];
  float* op = agg + (size_t)d * DH + lane * 4;
  atomicAdd(op + 0, hv.x * alpha);
  atomicAdd(op + 1, hv.y * alpha);
  atomicAdd(op + 2, hv.z * alpha);
  atomicAdd(op + 3, hv.w * alpha);
}

// ---------- bias + ReLU (in place on agg) ----------
__global__ void bias_relu_kernel(float* __restrict__ agg, const float* __restrict__ b, int n) {
  int i = blockIdx.x * blockDim.x + threadIdx.x;
  if (i >= n * DH) return;
  float v = agg[i] + b[i & (DH - 1)];
  agg[i] = v > 0.f ? v : 0.f;
}

// ---------- global add pool over batch ----------
__global__ void pool_kernel(const float* __restrict__ x, const int* __restrict__ batch,
                            float* __restrict__ pooled, int n) {
  int t = blockIdx.x * blockDim.x + threadIdx.x;
  int node = t >> 5;
  if (node >= n) return;
  int lane = t & 31;
  int g = batch[node];
  float4 v = ((const float4*)(x + (size_t)node * DH))[lane];
  float* op = pooled + (size_t)g * DH + lane * 4;
  atomicAdd(op + 0, v.x);
  atomicAdd(op + 1, v.y);
  atomicAdd(op + 2, v.z);
  atomicAdd(op + 3, v.w);
}

// ---------- final linear: y[g] = <pooled[g], wf> + bf ----------
__global__ __launch_bounds__(128)
void final_kernel(const float* __restrict__ pooled, const float* __restrict__ wf,
                  const float* __restrict__ bf, float* __restrict__ y) {
  __shared__ float red[128];
  int g = blockIdx.x, t = threadIdx.x;
  red[t] = pooled[(size_t)g * DH + t] * wf[t];
  __syncthreads();
  #pragma unroll
  for (int s = 64; s > 0; s >>= 1) {
    if (t < s) red[t] += red[t + s];
    __syncthreads();
  }
  if (t == 0) y[g] = red[0] + bf[0];
}

extern "C" void kernel_launch(void* const* d_in, const int* in_sizes, int n_in,
                              void* d_out, int out_size, void* d_ws, size_t ws_size,
                              hipStream_t stream) {
  const float* x      = (const float*)d_in[0];
  const int*   ei     = (const int*)d_in[1];
  const int*   batch  = (const int*)d_in[2];
  const float* Wm[3]  = {(const float*)d_in[3], (const float*)d_in[7],  (const float*)d_in[11]};
  const float* As[3]  = {(const float*)d_in[4], (const float*)d_in[8],  (const float*)d_in[12]};
  const float* Ad[3]  = {(const float*)d_in[5], (const float*)d_in[9],  (const float*)d_in[13]};
  const float* Bb[3]  = {(const float*)d_in[6], (const float*)d_in[10], (const float*)d_in[14]};
  const float* wf     = (const float*)d_in[15];
  const float* bf     = (const float*)d_in[16];
  float*       y      = (float*)d_out;

  // workspace carve-up
  float*    h      = (float*)d_ws;                       // N*DH
  float*    agg    = h    + (size_t)N_NODES * DH;        // N*DH
  float*    ssrc   = agg  + (size_t)N_NODES * DH;        // N
  float*    sdst   = ssrc + N_NODES;                     // N
  float*    e      = sdst + N_NODES;                     // EP
  unsigned* mEnc   = (unsigned*)(e + EP);                // N
  float*    denom  = (float*)(mEnc + N_NODES);           // N
  float*    pooled = denom + N_NODES;                    // NGRAPH*DH

  const unsigned NEG_INF_ORD = ~0xFF800000u;             // f2ord(-inf)

  dim3 gemmGrid((N_NODES + 127) / 128, DH / 16);
  int scoreBlocks = (N_NODES + 7) / 8;
  int edgeBlocks  = (EP + 255) / 256;
  int aggrBlocks  = (EP + 7) / 8;
  int nodeFeatBlk = (N_NODES * DH + 255) / 256;

  const float* cur = x;
  for (int l = 0; l < 3; ++l) {
    gemm_wmma_kernel<<<gemmGrid, 256, 0, stream>>>(cur, Wm[l], h, N_NODES);
    score_kernel<<<scoreBlocks, 256, 0, stream>>>(h, As[l], Ad[l], ssrc, sdst, N_NODES);

    fill_u32_kernel<<<(N_NODES + 255) / 256, 256, 0, stream>>>(mEnc, NEG_INF_ORD, N_NODES);
    fill_u32_kernel<<<(N_NODES + 255) / 256, 256, 0, stream>>>((unsigned*)denom, 0u, N_NODES);
    fill_u32_kernel<<<nodeFeatBlk, 256, 0, stream>>>((unsigned*)agg, 0u, N_NODES * DH);

    edge_max_kernel<<<edgeBlocks, 256, 0, stream>>>(ei, ssrc, sdst, e, mEnc);
    edge_exp_kernel<<<edgeBlocks, 256, 0, stream>>>(ei, e, mEnc, denom);
    edge_aggr_kernel<<<aggrBlocks, 256, 0, stream>>>(ei, e, denom, h, agg);
    bias_relu_kernel<<<nodeFeatBlk, 256, 0, stream>>>(agg, Bb[l], N_NODES);
    cur = agg;   // next layer input (h buffer is rewritten by next GEMM)
  }

  fill_u32_kernel<<<(NGRAPH * DH + 255) / 256, 256, 0, stream>>>((unsigned*)pooled, 0u, NGRAPH * DH);
  pool_kernel<<<(N_NODES * 32 + 255) / 256, 256, 0, stream>>>(cur, batch, pooled, N_NODES);
  final_kernel<<<NGRAPH, 128, 0, stream>>>(pooled, wf, bf, y);
}